// FastLocalAttention__86509231276510
// MI455X (gfx1250) — compile-verified
//
#include <hip/hip_runtime.h>

// Performer (FAVOR+) causal linear attention for MI455X (gfx1250).
// b=2,h=8,n=2048,d=64,m=256.  Chunk-parallel scan: per-chunk partial sums
// (WMMA f32 16x16x4), exclusive prefix over chunks, then per-chunk output.

#define BH    16            // b*h
#define SEQ   2048
#define DH    64
#define MF    256
#define CH    64            // chunk length
#define NCH   (SEQ / CH)    // 32 chunks per head
#define NSEG  (BH * NCH)    // 512 independent chunk-segments
#define LDK   260           // padded LDS stride for 256-wide rows
#define LDG   68            // padded LDS stride for 64-wide rows

#define NORMALIZER 0.35355339059327379f  // 64^-0.25
#define DIAGC      0.0625f               // 0.5 * 64^-0.5
#define RATIO      0.0625f               // 256^-0.5
#define EPSK       1e-4f
#define EPSD       1e-6f

typedef __attribute__((ext_vector_type(2))) float v2f;
typedef __attribute__((ext_vector_type(8))) float v8f;

__device__ __forceinline__ v8f wmma4(v2f a, v2f b, v8f c) {
  // D = A(16x4,f32) * B(4x16,f32) + C(16x16,f32)
  return __builtin_amdgcn_wmma_f32_16x16x4_f32(false, a, false, b, (short)0, c,
                                               false, false);
}

// ---- fragment loaders (wave32 layouts per CDNA5 ISA 7.12.2) -------------
// A 16x4: lanes0-15 rows, VGPR0/1 = K,K+1 ; lanes16-31 = K+2,K+3
__device__ __forceinline__ v2f ldA_g(const float* p, int row0, int kb, int lane,
                                     float scale) {
  int r = row0 + (lane & 15);
  int k = kb + ((lane >> 4) << 1);
  const float* s = p + (size_t)r * DH + k;
  v2f a; a.x = scale * s[0]; a.y = scale * s[1]; return a;
}
__device__ __forceinline__ v2f ldA_rm(const float* lds, int ld, int row0, int kb,
                                      int lane) {
  int r = row0 + (lane & 15);
  int k = kb + ((lane >> 4) << 1);
  v2f a; a.x = lds[r * ld + k]; a.y = lds[r * ld + k + 1]; return a;
}
__device__ __forceinline__ v2f ldA_t(const float* lds, int ld, int row0, int kb,
                                     int lane) {  // A[r][k] = lds[k*ld + r]
  int r = row0 + (lane & 15);
  int k = kb + ((lane >> 4) << 1);
  v2f a; a.x = lds[k * ld + r]; a.y = lds[(k + 1) * ld + r]; return a;
}
// B 4x16: lanes0-15 cols, VGPR0/1 = K,K+1 ; lanes16-31 = K+2,K+3
__device__ __forceinline__ v2f ldB_rm(const float* src, int ld, int n0, int kb,
                                      int lane) {
  int n = n0 + (lane & 15);
  int k = kb + ((lane >> 4) << 1);
  v2f b; b.x = src[(size_t)k * ld + n]; b.y = src[(size_t)(k + 1) * ld + n];
  return b;
}
__device__ __forceinline__ v2f ldB_t(const float* lds, int ld, int n0, int kb,
                                     int lane) {  // B[k][n] = lds[n*ld + k]
  int n = n0 + (lane & 15);
  int k = kb + ((lane >> 4) << 1);
  v2f b; b.x = lds[n * ld + k]; b.y = lds[n * ld + k + 1]; return b;
}
__device__ __forceinline__ v2f ldB_proj(const float* proj, int m0, int kb,
                                        int lane) {  // B[k][n] = proj[n][k]
  int n = m0 + (lane & 15);
  int k = kb + ((lane >> 4) << 1);
  v2f b; b.x = proj[n * DH + k]; b.y = proj[n * DH + k + 1]; return b;
}
// C/D 16x16 f32: VGPR i -> M = i + (lane>=16 ? 8:0), N = lane&15

// ---- ordered-uint float max encoding ------------------------------------
__device__ __forceinline__ unsigned f2ord(float f) {
  unsigned u = __float_as_uint(f);
  return (u & 0x80000000u) ? ~u : (u | 0x80000000u);
}
__device__ __forceinline__ float ord2f(unsigned u) {
  return (u & 0x80000000u) ? __uint_as_float(u & 0x7fffffffu)
                           : __uint_as_float(~u);
}

// ---- phi feature map: feat[CH][LDK] = ratio*(exp(xd - diag - mx) + eps) --
__device__ void compute_phi(const float* xbase, const float* proj, float* feat,
                            float* diag, float* rmax, bool is_query,
                            float headmax, int tid) {
  int wave = tid >> 5, lane = tid & 31;
  // xd = (normalizer * X) . P^T   -- 4x16 tiles of 16x16, K=64
  for (int t = wave; t < (CH / 16) * (MF / 16); t += 8) {
    int rt = t >> 4, ct = t & 15;
    v8f c = {};
    for (int kb = 0; kb < DH; kb += 4) {
      v2f a = ldA_g(xbase, rt * 16, kb, lane, NORMALIZER);
      v2f b = ldB_proj(proj, ct * 16, kb, lane);
      c = wmma4(a, b, c);
    }
    int n = ct * 16 + (lane & 15);
    int mb = rt * 16 + (lane >> 4) * 8;
    for (int i = 0; i < 8; i++) feat[(mb + i) * LDK + n] = c[i];
  }
  __syncthreads();
  if (tid < CH) {
    float s = 0.f;
    const float* xr = xbase + (size_t)tid * DH;
    for (int k = 0; k < DH; k++) s += xr[k] * xr[k];
    diag[tid] = DIAGC * s;
    if (is_query) {
      float mx = -__builtin_inff();
      const float* fr = feat + tid * LDK;
      for (int m = 0; m < MF; m++) mx = fmaxf(mx, fr[m]);
      rmax[tid] = mx;
    }
  }
  __syncthreads();
  for (int idx = tid; idx < CH * MF; idx += 256) {
    int r = idx >> 8, m = idx & 255;
    float mx = is_query ? rmax[r] : headmax;
    float xv = feat[r * LDK + m];
    feat[r * LDK + m] = RATIO * (__expf(xv - diag[r] - mx) + EPSK);
  }
  __syncthreads();
}

// ---- kernel 0: init per-head key-max slots ------------------------------
__global__ void init_kernel(unsigned* kmaxu) {
  if (threadIdx.x < BH) kmaxu[threadIdx.x] = f2ord(-__builtin_inff());
}

// ---- kernel 1: per-head max of key projection ---------------------------
__global__ __launch_bounds__(256) void kmax_kernel(const float* kin,
                                                   const float* proj,
                                                   unsigned* kmaxu) {
  __shared__ float red[256];
  int seg = blockIdx.x;
  int head = seg / NCH, chunk = seg % NCH;
  const float* xbase = kin + (size_t)(head * SEQ + chunk * CH) * DH;
  int tid = threadIdx.x, wave = tid >> 5, lane = tid & 31;
  float lmax = -__builtin_inff();
  for (int t = wave; t < 64; t += 8) {
    int rt = t >> 4, ct = t & 15;
    v8f c = {};
    for (int kb = 0; kb < DH; kb += 4) {
      v2f a = ldA_g(xbase, rt * 16, kb, lane, NORMALIZER);
      v2f b = ldB_proj(proj, ct * 16, kb, lane);
      c = wmma4(a, b, c);
    }
    for (int i = 0; i < 8; i++) lmax = fmaxf(lmax, c[i]);
  }
  red[tid] = lmax;
  __syncthreads();
  for (int s = 128; s > 0; s >>= 1) {
    if (tid < s) red[tid] = fmaxf(red[tid], red[tid + s]);
    __syncthreads();
  }
  if (tid == 0) atomicMax(&kmaxu[head], f2ord(red[0]));
}

// ---- kernel 2: per-chunk partials  S_c = K'^T V,  z_c = colsum(K') ------
__global__ __launch_bounds__(256) void chunk_sums_kernel(
    const float* kin, const float* v, const float* proj, float* Spart,
    float* zpart, const unsigned* kmaxu) {
  __shared__ float kp[CH * LDK];
  __shared__ float diag[CH];
  int seg = blockIdx.x;
  int head = seg / NCH, chunk = seg % NCH;
  const float* kbase = kin + (size_t)(head * SEQ + chunk * CH) * DH;
  const float* vbase = v + (size_t)(head * SEQ + chunk * CH) * DH;
  int tid = threadIdx.x, wave = tid >> 5, lane = tid & 31;
  float hmax = ord2f(kmaxu[head]);
  compute_phi(kbase, proj, kp, diag, diag, false, hmax, tid);
  // S_c = K'^T V : [256x64], 16x4 tiles, K = 64 chunk rows
  float* S = Spart + (size_t)seg * (MF * DH);
  for (int t = wave; t < (MF / 16) * (DH / 16); t += 8) {
    int rt = t >> 2, ct = t & 3;
    v8f c = {};
    for (int kb = 0; kb < CH; kb += 4) {
      v2f a = ldA_t(kp, LDK, rt * 16, kb, lane);    // A[m][r] = kp[r][m]
      v2f b = ldB_rm(vbase, DH, ct * 16, kb, lane); // B[r][d]
      c = wmma4(a, b, c);
    }
    int n = ct * 16 + (lane & 15);
    int mb = rt * 16 + (lane >> 4) * 8;
    for (int i = 0; i < 8; i++) S[(size_t)(mb + i) * DH + n] = c[i];
  }
  if (tid < MF) {
    float s = 0.f;
    for (int r = 0; r < CH; r++) s += kp[r * LDK + tid];
    zpart[(size_t)seg * MF + tid] = s;
  }
}

// ---- kernel 3: exclusive prefix over chunks, in place -------------------
__global__ void prefix_kernel(float* Spart, float* zpart) {
  const int per = MF * DH + MF;  // 16640 state entries per head
  int gid = blockIdx.x * blockDim.x + threadIdx.x;
  int head = gid / per, e = gid % per;
  if (head >= BH) return;
  if (e < MF * DH) {
    float run = 0.f;
    for (int c = 0; c < NCH; c++) {
      size_t idx = (size_t)(head * NCH + c) * (MF * DH) + e;
      float t = Spart[idx]; Spart[idx] = run; run += t;
    }
  } else {
    int m = e - MF * DH;
    float run = 0.f;
    for (int c = 0; c < NCH; c++) {
      size_t idx = (size_t)(head * NCH + c) * MF + m;
      float t = zpart[idx]; zpart[idx] = run; run += t;
    }
  }
}

// ---- kernel 4: out = (Q' S_prev + tril(Q'K'^T) V) / den -----------------
__global__ __launch_bounds__(256) void output_kernel(
    const float* q, const float* kin, const float* v, const float* proj,
    const float* Spart, const float* zpart, const unsigned* kmaxu,
    float* out) {
  __shared__ float qp[CH * LDK];
  __shared__ float kp[CH * LDK];
  __shared__ float g[CH * LDG];
  __shared__ float diag[CH], rmax[CH], dinv[CH];
  int seg = blockIdx.x;
  int head = seg / NCH, chunk = seg % NCH;
  size_t rowoff = (size_t)(head * SEQ + chunk * CH) * DH;
  const float* qbase = q + rowoff;
  const float* kbase = kin + rowoff;
  const float* vbase = v + rowoff;
  const float* Sprev = Spart + (size_t)seg * (MF * DH);
  const float* zprev = zpart + (size_t)seg * MF;
  int tid = threadIdx.x, wave = tid >> 5, lane = tid & 31;
  float hmax = ord2f(kmaxu[head]);
  compute_phi(qbase, proj, qp, diag, rmax, true, 0.f, tid);
  compute_phi(kbase, proj, kp, diag, rmax, false, hmax, tid);
  // G = Q' K'^T with causal (lower-tri incl. diagonal) mask; 4x4 tiles, K=256
  for (int t = wave; t < 16; t += 8) {
    int rt = t >> 2, ct = t & 3;
    v8f c = {};
    for (int kb = 0; kb < MF; kb += 4) {
      v2f a = ldA_rm(qp, LDK, rt * 16, kb, lane);
      v2f b = ldB_t(kp, LDK, ct * 16, kb, lane);  // B[m][j] = kp[j][m]
      c = wmma4(a, b, c);
    }
    int n = ct * 16 + (lane & 15);
    int mb = rt * 16 + (lane >> 4) * 8;
    for (int i = 0; i < 8; i++) {
      int row = mb + i;
      g[row * LDG + n] = (n <= row) ? c[i] : 0.f;
    }
  }
  __syncthreads();
  // denominator: q'.z_prev + cum-rowsum(tril G) + eps * rowsum(q')
  if (tid < CH) {
    float s = 0.f, qs = 0.f;
    for (int m = 0; m < MF; m++) {
      float qv = qp[tid * LDK + m];
      s += qv * zprev[m];
      qs += qv;
    }
    float intra = 0.f;
    for (int j = 0; j <= tid; j++) intra += g[tid * LDG + j];
    dinv[tid] = 1.f / (s + intra + EPSD * qs);
  }
  __syncthreads();
  // out tiles: K=256 against S_prev then K=64 against V, scaled by dinv
  for (int t = wave; t < 16; t += 8) {
    int rt = t >> 2, ct = t & 3;
    v8f c = {};
    for (int kb = 0; kb < MF; kb += 4) {
      v2f a = ldA_rm(qp, LDK, rt * 16, kb, lane);
      v2f b = ldB_rm(Sprev, DH, ct * 16, kb, lane);
      c = wmma4(a, b, c);
    }
    for (int kb = 0; kb < CH; kb += 4) {
      v2f a = ldA_rm(g, LDG, rt * 16, kb, lane);
      v2f b = ldB_rm(vbase, DH, ct * 16, kb, lane);
      c = wmma4(a, b, c);
    }
    int n = ct * 16 + (lane & 15);
    int mb = rt * 16 + (lane >> 4) * 8;
    for (int i = 0; i < 8; i++) {
      int row = mb + i;
      out[rowoff + (size_t)row * DH + ct * 16 + (lane & 15)] = c[i] * dinv[row];
    }
  }
}

extern "C" void kernel_launch(void* const* d_in, const int* in_sizes, int n_in,
                              void* d_out, int out_size, void* d_ws,
                              size_t ws_size, hipStream_t stream) {
  const float* q = (const float*)d_in[0];
  const float* k = (const float*)d_in[1];
  const float* v = (const float*)d_in[2];
  const float* proj = (const float*)d_in[3];
  float* out = (float*)d_out;

  float* Spart = (float*)d_ws;                       // NSEG * 256 * 64
  float* zpart = Spart + (size_t)NSEG * MF * DH;     // NSEG * 256
  unsigned* kmaxu = (unsigned*)(zpart + (size_t)NSEG * MF);  // BH

  init_kernel<<<1, 32, 0, stream>>>(kmaxu);
  kmax_kernel<<<NSEG, 256, 0, stream>>>(k, proj, kmaxu);
  chunk_sums_kernel<<<NSEG, 256, 0, stream>>>(k, v, proj, Spart, zpart, kmaxu);
  const int per = MF * DH + MF;
  prefix_kernel<<<(BH * per + 255) / 256, 256, 0, stream>>>(Spart, zpart);
  output_kernel<<<NSEG, 256, 0, stream>>>(q, k, v, proj, Spart, zpart, kmaxu,
                                          out);
}